// TCAContextEmbedding_72507637891620
// MI455X (gfx1250) — compile-verified
//
#include <hip/hip_runtime.h>

// ---------------------------------------------------------------------------
// TCAContextEmbedding unfold: out[b,l,k,:] = x[b, l+off[k], :] (0 if OOB)
// offsets (CONTEXT_SIZE=9): [-3,-2,-1,0,0,0,1,2,3]
// x: [16,4096,128] f32 (32 MB), out: [16,4096,1152] f32 (288 MB).
// Pure bandwidth problem (~320 MB HBM traffic -> ~14us at 23.3 TB/s).
// CDNA5 path: async global<->LDS DMA (ASYNCcnt), VGPR-free data movement.
// Stores use TH=NT (non-temporal) so the write-once 288 MB output stream
// does not evict the reusable 32 MB input from the 192 MB L2.
// ---------------------------------------------------------------------------

#if defined(__AMDGCN__)
typedef int v4i __attribute__((vector_size(16)));
typedef __attribute__((address_space(1))) v4i as1_v4i;   // global ("__device__")
typedef __attribute__((address_space(3))) v4i as3_v4i;   // LDS
typedef __attribute__((address_space(3))) void as3_void;
#endif

#if defined(__AMDGCN__) && __has_builtin(__builtin_amdgcn_global_load_async_to_lds_b128)
#define HAVE_ASYNC_LOAD_BUILTIN 1
#endif
#if defined(__AMDGCN__) && __has_builtin(__builtin_amdgcn_global_store_async_from_lds_b128)
#define HAVE_ASYNC_STORE_BUILTIN 1
#endif

// Cache-policy immediates (gfx12 CPol: TH in [2:0]).  Load TH=0 (RT),
// Store TH=1 (NT) per CDNA5 ISA temporal-hint tables.
#define CPOL_LOAD_RT 0
#define CPOL_STORE_NT 1

// Async copy 16B: global -> LDS (tracked by ASYNCcnt; data bypasses VGPRs)
__device__ __forceinline__ void async_g2l_b128(const float* g, float* lds) {
#if defined(__AMDGCN__)
#if defined(HAVE_ASYNC_LOAD_BUILTIN)
  __builtin_amdgcn_global_load_async_to_lds_b128(
      (as1_v4i*)g, (as3_v4i*)lds, /*offset=*/0, /*cpol=*/CPOL_LOAD_RT);
#else
  unsigned lds_addr = (unsigned)(unsigned long long)(as3_void*)lds;
  asm volatile("global_load_async_to_lds_b128 %0, %1, off"
               :
               : "v"(lds_addr), "v"(g)
               : "memory");
#endif
#else
  (void)g; (void)lds;
#endif
}

// Async copy 16B: LDS -> global, non-temporal (streaming output)
__device__ __forceinline__ void async_l2g_b128_nt(float* g, const float* lds) {
#if defined(__AMDGCN__)
#if defined(HAVE_ASYNC_STORE_BUILTIN)
  __builtin_amdgcn_global_store_async_from_lds_b128(
      (as1_v4i*)g, (as3_v4i*)lds, /*offset=*/0, /*cpol=*/CPOL_STORE_NT);
#else
  unsigned lds_addr = (unsigned)(unsigned long long)(as3_void*)lds;
  asm volatile("global_store_async_from_lds_b128 %1, %0, off th:TH_STORE_NT"
               :
               : "v"(lds_addr), "v"(g)
               : "memory");
#endif
#else
  (void)g; (void)lds;
#endif
}

__device__ __forceinline__ void wait_async0() {
#if defined(__AMDGCN__)
#if __has_builtin(__builtin_amdgcn_s_wait_asynccnt)
  __builtin_amdgcn_s_wait_asynccnt(0);
#else
  asm volatile("s_wait_asynccnt 0" ::: "memory");
#endif
#endif
}

// Problem constants
constexpr int Bc   = 16;
constexpr int Lc   = 4096;
constexpr int Dc   = 128;   // embed dim
constexpr int Kc   = 9;     // context size
constexpr int PADc = 3;     // max |offset|
constexpr int SEG  = 64;    // seq rows per block
constexpr int ROWS = SEG + 2 * PADc;      // 70 rows staged in LDS
constexpr int NTHREADS = 256;             // 8 waves (wave32)

// offsets[k] for k=0..8 -> {-3,-2,-1,0, 0, 0, 1,2,3}
__device__ __forceinline__ int ctx_off(int k) {
  return (k < 4) ? (k - 3) : ((k < 6) ? 0 : (k - 5));
}

__global__ __launch_bounds__(NTHREADS)
void tca_context_unfold(const float* __restrict__ x, float* __restrict__ out) {
  __shared__ __align__(16) float tile[ROWS * Dc];   // 35,840 B

  const int tid = threadIdx.x;
  const int blocksPerBatch = Lc / SEG;              // 64
  const int b  = blockIdx.x / blocksPerBatch;
  const int s0 = (blockIdx.x % blocksPerBatch) * SEG;
  const float* xb = x + (size_t)b * Lc * Dc;

  // ---- Stage 1: async DMA the (SEG+6) x 128 tile into LDS ----------------
  // Main body: rows [s0, s0+SEG) are always in range. 2048 x b128 copies,
  // 8 per thread; each instruction moves a 512B contiguous burst per wave.
#pragma unroll
  for (int it = 0; it < (SEG * (Dc / 4)) / NTHREADS; ++it) {
    const int i = it * NTHREADS + tid;
    const int r = i >> 5;              // Dc/4 == 32 vec4 per row
    const int c = (i & 31) << 2;       // float offset within row
    async_g2l_b128(xb + (size_t)(s0 + r) * Dc + c,
                   &tile[(r + PADc) * Dc + c]);
  }
  // Halo: 3 rows before + 3 rows after (192 x b128). OOB rows -> DS zero-fill
  // (disjoint LDS addresses vs async writes, so no DS<->ASYNC ordering hazard).
  if (tid < 2 * PADc * (Dc / 4)) {
    const int h  = tid >> 5;
    const int c  = (tid & 31) << 2;
    const int gl = (h < PADc) ? (s0 - PADc + h) : (s0 + SEG + h - PADc);
    const int lr = (h < PADc) ? h : (h + SEG);
    float* ldst = &tile[lr * Dc + c];
    if ((unsigned)gl < (unsigned)Lc) {
      async_g2l_b128(xb + (size_t)gl * Dc + c, ldst);
    } else {
      float4 z = make_float4(0.f, 0.f, 0.f, 0.f);
      *(float4*)ldst = z;
    }
  }

  wait_async0();      // this wave's async loads have landed in LDS
  __syncthreads();    // + DS zero-fills visible; all waves' tiles complete

  // ---- Stage 2: async DMA the 9 shifted copies from LDS to global --------
  // out row layout [l][k][d] is fully contiguous, so consecutive i across the
  // 32 lanes of a wave -> one 512B contiguous burst per store instruction.
  float* ob = out + ((size_t)b * Lc + s0) * (size_t)(Kc * Dc);
  constexpr int F4_PER_ROW = Kc * (Dc / 4);              // 288
#pragma unroll 8
  for (int it = 0; it < (SEG * F4_PER_ROW) / NTHREADS; ++it) {   // 72 iters
    const int i   = it * NTHREADS + tid;
    const int l   = i / F4_PER_ROW;
    const int rem = i - l * F4_PER_ROW;
    const int k   = rem >> 5;
    const int c   = (rem & 31) << 2;
    const int srow = l + PADc + ctx_off(k);
    async_l2g_b128_nt(ob + (size_t)l * (Kc * Dc) + k * Dc + c,
                      &tile[srow * Dc + c]);
  }
  // S_ENDPGM performs an implicit S_WAIT_IDLE, draining outstanding async
  // stores before the wave (and its LDS) is retired.
}

extern "C" void kernel_launch(void* const* d_in, const int* in_sizes, int n_in,
                              void* d_out, int out_size, void* d_ws, size_t ws_size,
                              hipStream_t stream) {
  (void)in_sizes; (void)n_in; (void)out_size; (void)d_ws; (void)ws_size;
  const float* x = (const float*)d_in[0];
  float* out = (float*)d_out;
  const int nblocks = Bc * (Lc / SEG);   // 16 * 64 = 1024
  tca_context_unfold<<<nblocks, NTHREADS, 0, stream>>>(x, out);
}